// MixtureOfExperts_1769526526605
// MI455X (gfx1250) — compile-verified
//
#include <hip/hip_runtime.h>

typedef __attribute__((ext_vector_type(2))) float v2f;
typedef __attribute__((ext_vector_type(8))) float v8f;

#define D_MODEL   1024
#define D_FF      4096
#define NEXPERTS  8
#define NTOK      8192           // 4 * 2048
#define TOPK      2
#define XLDA      1028           // 1024 + 4 pad -> conflict-free fragment reads
#define HLDA      132            // 128 + 4 pad
#define TILES_PER_EXPERT 512     // ceil(NTOK/16) worst case

// --- CDNA5 async global->LDS copy (ASYNCcnt-tracked, bypasses VGPRs) -------
__device__ __forceinline__ void async_g2l_b128(unsigned lds_byte_off,
                                               const void* gaddr) {
    asm volatile("global_load_async_to_lds_b128 %0, %1, off"
                 :: "v"(lds_byte_off), "v"(gaddr)
                 : "memory");
}
__device__ __forceinline__ void wait_async0() {
#if __has_builtin(__builtin_amdgcn_s_wait_asynccnt)
    __builtin_amdgcn_s_wait_asynccnt(0);
#else
    asm volatile("s_wait_asynccnt 0" ::: "memory");
#endif
}

// ---------------------------------------------------------------------------
// Kernel 1: router. One wave (32 lanes) per token.
// ---------------------------------------------------------------------------
__global__ __launch_bounds__(256) void moe_router(
    const float* __restrict__ x, const float* __restrict__ gate_w,
    const float* __restrict__ gate_b,
    int* __restrict__ counts, float* __restrict__ probsum,
    int* __restrict__ list, float* __restrict__ wlist)
{
    __shared__ float sProb[NEXPERTS];
    const int tid  = threadIdx.x;
    const int wave = tid >> 5;
    const int lane = tid & 31;
    if (tid < NEXPERTS) sProb[tid] = 0.f;
    __syncthreads();

    const int token = blockIdx.x * 8 + wave;       // grid = 1024 blocks
    float acc[NEXPERTS];
#pragma unroll
    for (int e = 0; e < NEXPERTS; ++e) acc[e] = 0.f;

    const float* xr = x + (size_t)token * D_MODEL;
    for (int d = lane; d < D_MODEL; d += 32) {
        float xv = xr[d];
        const float* g = gate_w + (size_t)d * NEXPERTS;
#pragma unroll
        for (int e = 0; e < NEXPERTS; ++e) acc[e] += xv * g[e];
    }
#pragma unroll
    for (int off = 16; off > 0; off >>= 1) {
#pragma unroll
        for (int e = 0; e < NEXPERTS; ++e)
            acc[e] += __shfl_xor(acc[e], off, 32);
    }
#pragma unroll
    for (int e = 0; e < NEXPERTS; ++e) acc[e] += gate_b[e];

    // softmax over 8 experts (all lanes redundantly)
    float m = acc[0];
#pragma unroll
    for (int e = 1; e < NEXPERTS; ++e) m = fmaxf(m, acc[e]);
    float p[NEXPERTS];
    float s = 0.f;
#pragma unroll
    for (int e = 0; e < NEXPERTS; ++e) { p[e] = __expf(acc[e] - m); s += p[e]; }
    float inv = 1.f / s;
#pragma unroll
    for (int e = 0; e < NEXPERTS; ++e) p[e] *= inv;

    // top-2, ties -> lower index (strict >)
    int i0 = 0; float p0 = p[0];
#pragma unroll
    for (int e = 1; e < NEXPERTS; ++e) if (p[e] > p0) { p0 = p[e]; i0 = e; }
    int i1 = (i0 == 0) ? 1 : 0; float p1 = p[i1];
#pragma unroll
    for (int e = 0; e < NEXPERTS; ++e)
        if (e != i0 && p[e] > p1) { p1 = p[e]; i1 = e; }
    float rn = 1.f / (p0 + p1);

    if (lane == 0) {
#pragma unroll
        for (int e = 0; e < NEXPERTS; ++e) atomicAdd(&sProb[e], p[e]);
        int pos0 = atomicAdd(&counts[i0], 1);
        list[i0 * NTOK + pos0]  = token;
        wlist[i0 * NTOK + pos0] = p0 * rn;
        int pos1 = atomicAdd(&counts[i1], 1);
        list[i1 * NTOK + pos1]  = token;
        wlist[i1 * NTOK + pos1] = p1 * rn;
    }
    __syncthreads();
    if (tid < NEXPERTS) atomicAdd(&probsum[tid], sProb[tid]);
}

// ---------------------------------------------------------------------------
// Kernel 2: fused gathered expert FFN with fp32 WMMA (16x16x4).
// Block = 256 threads = 8 waves. Tile: 16 gathered tokens x 1024 out cols.
// ---------------------------------------------------------------------------
__global__ __launch_bounds__(256) void moe_ffn(
    const float* __restrict__ x,
    const float* __restrict__ w1, const float* __restrict__ b1,
    const float* __restrict__ w2, const float* __restrict__ b2,
    const int* __restrict__ counts, const int* __restrict__ list,
    const float* __restrict__ wlist, float* __restrict__ out)
{
    __shared__ float sX[16 * XLDA];   // 16 tokens x D_MODEL (padded)
    __shared__ float sH[16 * HLDA];   // 16 tokens x 128 D_FF chunk (padded)
    __shared__ int   sTok[16];
    __shared__ float sWgt[16];

    const int e   = blockIdx.x >> 9;            // / TILES_PER_EXPERT
    const int t   = blockIdx.x & (TILES_PER_EXPERT - 1);
    const int cnt = counts[e];
    if (t * 16 >= cnt) return;                  // uniform across block

    const int tid = threadIdx.x;
    if (tid < 16) {
        int idx = t * 16 + tid;
        if (idx < cnt) { sTok[tid] = list[e * NTOK + idx]; sWgt[tid] = wlist[e * NTOK + idx]; }
        else           { sTok[tid] = 0;                    sWgt[tid] = 0.f; }
    }
    __syncthreads();

    // Stage gathered X rows into LDS with CDNA5 async global->LDS b128 copies.
    // 16 rows x 1024 floats = 4096 float4 transfers, 16 per thread, EXEC full.
    {
        const int nvec = (16 * D_MODEL) / 4;
        for (int i = tid; i < nvec; i += 256) {
            int r  = i >> 8;                  // 256 float4 per row
            int c4 = i & 255;
            unsigned lds_off = (unsigned)(size_t)&sX[r * XLDA + c4 * 4];
            const float* gsrc = x + (size_t)sTok[r] * D_MODEL + c4 * 4;
            async_g2l_b128(lds_off, gsrc);
        }
        wait_async0();
    }
    __syncthreads();

    const int wave  = tid >> 5;
    const int lane  = tid & 31;
    const int lhalf = lane >> 4;     // selects K pair / M upper half
    const int lmod  = lane & 15;

    const v8f vzero = {0.f, 0.f, 0.f, 0.f, 0.f, 0.f, 0.f, 0.f};
    v8f acc[8];
#pragma unroll
    for (int i = 0; i < 8; ++i) acc[i] = vzero;

    const float* W1e = w1 + (size_t)e * D_MODEL * D_FF;
    const float* W2e = w2 + (size_t)e * D_FF * D_MODEL;

    for (int fc = 0; fc < D_FF; fc += 128) {
        // Warm L2/WGP$ for the next chunk's streaming weight reads.
        if (fc + 128 < D_FF) {
            __builtin_prefetch(W1e + (size_t)(wave * 4 + (lane >> 3)) * D_FF
                                   + (fc + 128) + (lane & 7) * 16, 0, 0);
            __builtin_prefetch(W2e + (size_t)(fc + 128 + wave * 16 + (lane >> 1)) * D_MODEL
                                   + (lane & 1) * 512, 0, 0);
        }

        // ---- stage 1: H[16 x 16] per wave, K = 1024 over sX ----
        const int fcol = fc + wave * 16 + lmod;
        v8f h = vzero;
        for (int k0 = 0; k0 < D_MODEL; k0 += 4) {
            const int k = k0 + 2 * lhalf;
            v2f a; a.x = sX[lmod * XLDA + k];
                   a.y = sX[lmod * XLDA + k + 1];
            v2f b; b.x = W1e[(size_t)k * D_FF + fcol];
                   b.y = W1e[(size_t)(k + 1) * D_FF + fcol];
            h = __builtin_amdgcn_wmma_f32_16x16x4_f32(
                    false, a, false, b, (short)0, h, false, false);
        }
        const float bias = b1[e * D_FF + fcol];
#pragma unroll
        for (int r = 0; r < 8; ++r) {
            float v = h[r] + bias;
            h[r] = v > 0.f ? v : 0.f;
        }
        __syncthreads();   // prior stage-2 readers of sH are done
#pragma unroll
        for (int r = 0; r < 8; ++r)
            sH[(r + 8 * lhalf) * HLDA + wave * 16 + lmod] = h[r];
        __syncthreads();

        // ---- stage 2: acc[16 x 1024] += H[16 x 128] @ W2chunk[128 x 1024] ----
        for (int k0 = 0; k0 < 128; k0 += 4) {
            const int k = k0 + 2 * lhalf;
            v2f a; a.x = sH[lmod * HLDA + k];
                   a.y = sH[lmod * HLDA + k + 1];
            const size_t rowbase = (size_t)(fc + k) * D_MODEL;
#pragma unroll
            for (int ct = 0; ct < 8; ++ct) {
                const int ncol = wave * 128 + ct * 16 + lmod;
                v2f b; b.x = W2e[rowbase + ncol];
                       b.y = W2e[rowbase + D_MODEL + ncol];
                acc[ct] = __builtin_amdgcn_wmma_f32_16x16x4_f32(
                              false, a, false, b, (short)0, acc[ct], false, false);
            }
        }
    }

    // ---- weighted scatter (each token is in exactly TOPK expert lists) ----
#pragma unroll
    for (int ct = 0; ct < 8; ++ct) {
        const int ncol = wave * 128 + ct * 16 + lmod;
        const float bias2 = b2[e * D_MODEL + ncol];
#pragma unroll
        for (int r = 0; r < 8; ++r) {
            const int mrow = r + 8 * lhalf;
            if (t * 16 + mrow < cnt) {
                float v = sWgt[mrow] * (acc[ct][r] + bias2);
                atomicAdd(&out[(size_t)sTok[mrow] * D_MODEL + ncol], v);
            }
        }
    }
}

// ---------------------------------------------------------------------------
// Kernel 3: routing statistics
// ---------------------------------------------------------------------------
__global__ void moe_finalize(const int* __restrict__ counts,
                             const float* __restrict__ probsum,
                             float* __restrict__ out_tail)
{
    __shared__ float avg[NEXPERTS];
    const int tid = threadIdx.x;
    if (tid < NEXPERTS) {
        out_tail[tid] = (float)counts[tid] / (float)(NTOK * TOPK);
        avg[tid] = probsum[tid] / (float)NTOK;
    }
    __syncthreads();
    if (tid == 0) {
        float mean = 0.f;
        for (int e = 0; e < NEXPERTS; ++e) mean += avg[e];
        mean *= (1.f / NEXPERTS);
        float v = 0.f;
        for (int e = 0; e < NEXPERTS; ++e) {
            float d = avg[e] - mean;
            v += d * d;
        }
        out_tail[NEXPERTS] = v / (float)(NEXPERTS - 1);   // unbiased var
    }
}

// ---------------------------------------------------------------------------
extern "C" void kernel_launch(void* const* d_in, const int* in_sizes, int n_in,
                              void* d_out, int out_size, void* d_ws, size_t ws_size,
                              hipStream_t stream) {
    const float* x      = (const float*)d_in[0];
    const float* gate_w = (const float*)d_in[1];
    const float* gate_b = (const float*)d_in[2];
    const float* w1     = (const float*)d_in[3];
    const float* b1     = (const float*)d_in[4];
    const float* w2     = (const float*)d_in[5];
    const float* b2     = (const float*)d_in[6];
    float* out = (float*)d_out;

    // workspace layout
    int*   counts  = (int*)d_ws;                                    // 8 ints
    float* probsum = (float*)((char*)d_ws + 32);                    // 8 floats
    int*   list    = (int*)((char*)d_ws + 64);                      // E*N ints
    float* wlist   = (float*)((char*)d_ws + 64 +
                              (size_t)NEXPERTS * NTOK * sizeof(int));

    hipMemsetAsync(d_ws, 0, 64, stream);
    hipMemsetAsync(d_out, 0, (size_t)out_size * sizeof(float), stream);

    moe_router<<<NTOK / 8, 256, 0, stream>>>(x, gate_w, gate_b,
                                             counts, probsum, list, wlist);
    moe_ffn<<<NEXPERTS * TILES_PER_EXPERT, 256, 0, stream>>>(
        x, w1, b1, w2, b2, counts, list, wlist, out);
    moe_finalize<<<1, 32, 0, stream>>>(counts, probsum,
                                       out + (size_t)NTOK * D_MODEL);
}